// Sampler_489626271766
// MI455X (gfx1250) — compile-verified
//
#include <hip/hip_runtime.h>
#include <hip/hip_bf16.h>
#include <math.h>

#define B_  32
#define S_  8
#define D_  2048
#define V_  128000
#define SOFTCAP 30.0f

typedef __attribute__((ext_vector_type(2))) float v2f;
typedef __attribute__((ext_vector_type(8))) float v8f;

// Branch-free tanh: exact to fp32 ULP for |y| >= 10 (saturates), no EXEC
// divergence. Softcap argument is logit/30, bounded in practice.
__device__ inline float fast_tanh(float y) {
  y = fminf(fmaxf(y, -10.f), 10.f);
  float e = __expf(2.f * y);
  return (e - 1.f) / (e + 1.f);
}

// ---------------------------------------------------------------------------
// Kernel 1: logits = hs @ emb^T, softcap + temperature.
// Memory-bound: streams 1.05 GB of fp32 embedding once (~45us @ 23.3 TB/s).
// Native fp32 WMMA 16x16x4 (exact). Each wave: one 16-wide vocab tile, both
// 16-row M halves, with 4 accumulators (even/odd K) for XDL pipelining.
// ---------------------------------------------------------------------------
__global__ __launch_bounds__(256) void logits_kernel(
    const float* __restrict__ emb, const float* __restrict__ hidden,
    const float* __restrict__ temps, const int* __restrict__ pos,
    float* __restrict__ logits_out)
{
  const int wave = threadIdx.x >> 5;
  const int lane = threadIdx.x & 31;
  const int n0   = (blockIdx.x * 8 + wave) * 16;   // vocab tile base
  const int p    = pos[0];
  const int l15  = lane & 15;
  const int kh   = lane >> 4;                      // K-half within fragment

  v8f acc0a = {0.f,0.f,0.f,0.f,0.f,0.f,0.f,0.f};
  v8f acc0b = acc0a, acc1a = acc0a, acc1b = acc0a;

  // 32-bit A 16x4 layout: lanes0-15 M=lane {K=0,1}; lanes16-31 M=lane-16 {K=2,3}
  // B 4x16 mirrors with N across lanes.
  const float* pa0 = hidden + ((size_t)l15 * S_ + p) * D_ + 2 * kh;   // A rows 0..15
  const float* pa1 = pa0 + (size_t)16 * S_ * D_;                      // A rows 16..31
  const float* pb  = emb + (size_t)(n0 + l15) * D_ + 2 * kh;          // B rows (vocab)

  #pragma unroll 4
  for (int k = 0; k < D_; k += 8) {
    // even 4-K substep -> "a" accumulators
    {
      v2f a0 = *(const v2f*)(pa0);
      v2f a1 = *(const v2f*)(pa1);
      v2f bb = *(const v2f*)(pb);
      acc0a = __builtin_amdgcn_wmma_f32_16x16x4_f32(false, a0, false, bb,
                                                    (short)0, acc0a, false, false);
      acc1a = __builtin_amdgcn_wmma_f32_16x16x4_f32(false, a1, false, bb,
                                                    (short)0, acc1a, false, false);
    }
    // odd 4-K substep -> "b" accumulators (dependency distance = 4 WMMAs)
    {
      v2f a0 = *(const v2f*)(pa0 + 4);
      v2f a1 = *(const v2f*)(pa1 + 4);
      v2f bb = *(const v2f*)(pb + 4);
      acc0b = __builtin_amdgcn_wmma_f32_16x16x4_f32(false, a0, false, bb,
                                                    (short)0, acc0b, false, false);
      acc1b = __builtin_amdgcn_wmma_f32_16x16x4_f32(false, a1, false, bb,
                                                    (short)0, acc1b, false, false);
    }
    pa0 += 8; pa1 += 8; pb += 8;
  }

  v8f acc0 = acc0a + acc0b;
  v8f acc1 = acc1a + acc1b;

  // C/D 16x16 f32 layout: lanes0-15 N=lane, VGPR r -> M=r; lanes16-31 M=8+r.
  const int ncol  = n0 + l15;
  const int mbase = kh * 8;
  #pragma unroll
  for (int r = 0; r < 8; r++) {
    int m0 = mbase + r;
    float x0 = fast_tanh(acc0[r] * (1.0f / SOFTCAP)) * SOFTCAP / temps[m0];
    logits_out[(size_t)m0 * V_ + ncol] = x0;
    int m1 = 16 + mbase + r;
    float x1 = fast_tanh(acc1[r] * (1.0f / SOFTCAP)) * SOFTCAP / temps[m1];
    logits_out[(size_t)m1 * V_ + ncol] = x1;
  }
}

// ---------------------------------------------------------------------------
// Kernel 2: per-row softmax stats + top-64 selection + top-p/top-k sample.
// ---------------------------------------------------------------------------
#define NT 1024
#define STRIP (V_ / NT)   // 125

// Unique, sortable key: larger value wins, ties -> smaller index (stable argsort).
__device__ inline unsigned long long mkkey(float v, unsigned idx) {
  unsigned u = __float_as_uint(v);
  u = (u & 0x80000000u) ? ~u : (u | 0x80000000u);
  return ((unsigned long long)u << 32) | (unsigned long long)(0xFFFFFFFFu - idx);
}

__global__ __launch_bounds__(NT) void sample_kernel(
    const float* __restrict__ logits, const float* __restrict__ top_ps,
    const int* __restrict__ top_ks, float* __restrict__ out_tok)
{
  __shared__ float               red_f[NT];
  __shared__ unsigned long long  red_k[NT];
  __shared__ float               sel_v[64];
  __shared__ int                 sel_i[64];

  const int b   = blockIdx.x;
  const int tid = threadIdx.x;
  const float* row = logits + (size_t)b * V_;
  const int s0 = tid * STRIP;

  // pass 1: row max
  float lmax = -3.4e38f;
  for (int i = 0; i < STRIP; i++) lmax = fmaxf(lmax, row[s0 + i]);
  red_f[tid] = lmax; __syncthreads();
  for (int s = NT / 2; s > 0; s >>= 1) {
    if (tid < s) red_f[tid] = fmaxf(red_f[tid], red_f[tid + s]);
    __syncthreads();
  }
  const float rmax = red_f[0];
  __syncthreads();

  // pass 2: sum exp
  float lsum = 0.f;
  for (int i = 0; i < STRIP; i++) lsum += __expf(row[s0 + i] - rmax);
  red_f[tid] = lsum; __syncthreads();
  for (int s = NT / 2; s > 0; s >>= 1) {
    if (tid < s) red_f[tid] += red_f[tid + s];
    __syncthreads();
  }
  const float Z = red_f[0];
  __syncthreads();

  // top-64 selection: per-thread best key, only the winner rescans its strip.
  unsigned long long best = 0ull;
  for (int i = 0; i < STRIP; i++) {
    unsigned long long k = mkkey(row[s0 + i], (unsigned)(s0 + i));
    if (k > best) best = k;
  }
  for (int it = 0; it < 64; it++) {
    red_k[tid] = best; __syncthreads();
    for (int s = NT / 2; s > 0; s >>= 1) {
      if (tid < s && red_k[tid + s] > red_k[tid]) red_k[tid] = red_k[tid + s];
      __syncthreads();
    }
    const unsigned long long win = red_k[0];
    if (tid == 0) {
      unsigned idx = 0xFFFFFFFFu - (unsigned)win;
      unsigned up  = (unsigned)(win >> 32);
      unsigned u   = (up & 0x80000000u) ? (up ^ 0x80000000u) : ~up;
      sel_v[it] = __uint_as_float(u);
      sel_i[it] = (int)idx;
    }
    __syncthreads();
    if (best == win) {               // unique keys -> exactly one owner
      best = 0ull;
      for (int i = 0; i < STRIP; i++) {
        unsigned long long k = mkkey(row[s0 + i], (unsigned)(s0 + i));
        if (k < win && k > best) best = k;
      }
    }
    __syncthreads();
  }

  if (tid == 0) {
    const float tp = top_ps[b];
    const int   tk = top_ks[b];
    float kept[64];
    float cum = 0.f, ksum = 0.f;
    for (int i = 0; i < 64; i++) {
      float pr = __expf(sel_v[i] - rmax) / Z;
      bool masked = (cum > tp) || (i >= tk);   // exclusive cumsum > top_p
      cum += pr;
      float kp = masked ? 0.f : pr;
      kept[i] = kp;
      ksum += kp;
    }
    // deterministic per-row uniform (splitmix64); jax Threefry not reproducible
    unsigned long long x = ((unsigned long long)(b + 1)) * 0x9E3779B97F4A7C15ull
                           ^ 0xD1B54A32D192ED03ull;
    x ^= x >> 30; x *= 0xBF58476D1CE4E5B9ull;
    x ^= x >> 27; x *= 0x94D049BB133111EBull;
    x ^= x >> 31;
    float u = (float)(unsigned)(x >> 40) * (1.0f / 16777216.0f);
    float target = u * ksum;
    float acc = 0.f;
    int tok = sel_i[0];
    for (int i = 0; i < 64; i++) {
      if (kept[i] > 0.f) {
        acc += kept[i];
        tok = sel_i[i];
        if (acc >= target) break;
      }
    }
    out_tok[b] = (float)tok;
  }
}

// ---------------------------------------------------------------------------
extern "C" void kernel_launch(void* const* d_in, const int* in_sizes, int n_in,
                              void* d_out, int out_size, void* d_ws, size_t ws_size,
                              hipStream_t stream) {
  (void)in_sizes; (void)n_in; (void)d_ws; (void)ws_size; (void)out_size;
  const float* emb    = (const float*)d_in[0];  // [V, D]
  const float* hidden = (const float*)d_in[1];  // [B, S, D]
  const float* temps  = (const float*)d_in[2];  // [B]
  const float* top_ps = (const float*)d_in[3];  // [B]
  const int*   pos    = (const int*)d_in[4];    // [1]
  const int*   top_ks = (const int*)d_in[5];    // [B]

  float* out        = (float*)d_out;            // [B] tokens, then [B*V] logits
  float* logits_out = out + B_;

  dim3 g1(V_ / 128);                            // 8 waves * 16 vocab rows / block
  logits_kernel<<<g1, 256, 0, stream>>>(emb, hidden, temps, pos, logits_out);
  sample_kernel<<<B_, NT, 0, stream>>>(logits_out, top_ps, top_ks, out);
}